// MANN_65000035058326
// MI455X (gfx1250) — compile-verified
//
#include <hip/hip_runtime.h>
#include <math.h>

// ---------------------------------------------------------------------------
// MANN (LSTM controller + external memory) forward, fp32, CDNA5 / gfx1250.
// GEMMs use V_WMMA_F32_16X16X4_F32 (fp32-faithful WMMA path, wave32).
// Memory subsystem fused into one workgroup per batch element, sim in LDS.
// ---------------------------------------------------------------------------

#define T_STEPS 128
#define BATCH   64
#define IN_DIM  128
#define H_DIM   512
#define N_SLOTS 2048
#define W_DIM   64
#define OUT_DIM 128
#define R_HEADS 4
#define RW_DIM  (R_HEADS * W_DIM)      // 256
#define CTRL_IN (IN_DIM + RW_DIM)      // 384
#define G4H     (4 * H_DIM)            // 2048
#define KEY_DIM ((R_HEADS + 1) * W_DIM) // 320
#define EPSF    1e-8f

// workspace layout (float offsets)
#define OFF_H      0u
#define OFF_C      32768u                       // B*H
#define OFF_RV     65536u                       // B*RW
#define OFF_USAGE  81920u                       // B*N
#define OFF_RWSUM  212992u                      // B*N
#define OFF_GATES  344064u                      // B*4H
#define OFF_KEYS   475136u                      // B*KEY_DIM
#define OFF_M      495616u                      // B*N*W
#define M_ELEMS    8388608u
#define WS_TOTAL   (OFF_M + M_ELEMS)            // ~35.5 MB (< harness ws)

typedef __attribute__((ext_vector_type(2))) float v2f;
typedef __attribute__((ext_vector_type(8))) float v8f;

__device__ __forceinline__ float sigmoidf(float x) { return 1.f / (1.f + expf(-x)); }

// ---------------------------------------------------------------------------
// fp32 WMMA 16x16x4 micro-GEMM over one K-segment.
// A: (rows of the 16-row batch tile) row-major, stride aStride.
// W: weights row-major (outUnit, k); B-fragment element (K,N=j) = W[j][k].
// Fragment layout per ISA 7.12.2: lanes 0-15 carry K={k,k+1}, lanes 16-31
// K={k+2,k+3} (8-byte contiguous loads for both A and B operands).
// 4 interleaved accumulators = 4 independent WMMA dependency chains.
// ---------------------------------------------------------------------------
__device__ __forceinline__ void gemm_seg(const float* __restrict__ A, int aStride,
                                         const float* __restrict__ W, int wStride, int wOff,
                                         int segK, int aRow, int wRow, int lhalf,
                                         v8f acc[4]) {
  const float* ap = A + (size_t)aRow * aStride;
  const float* wp = W + (size_t)wRow * wStride + wOff;
  for (int k0 = 0; k0 < segK; k0 += 16) {
#pragma unroll
    for (int u = 0; u < 4; ++u) {
      const int kl = k0 + 4 * u + 2 * lhalf;
      v2f a = *(const v2f*)(ap + kl);
      v2f b = *(const v2f*)(wp + kl);
      acc[u] = __builtin_amdgcn_wmma_f32_16x16x4_f32(false, a, false, b,
                                                     (short)0, acc[u], false, false);
    }
  }
}

// ---------------------------------------------------------------------------
// Kernel 1: gates = [x_t | read_vec | h] @ [Wih | Whh]^T + bih + bhh
// block = (32,4): 4 waves, each wave one 16x16 output tile. grid = (32,4).
// ---------------------------------------------------------------------------
__global__ __launch_bounds__(128) void k_gates(const float* __restrict__ x_seq,
                                               const float* __restrict__ rv,
                                               const float* __restrict__ h,
                                               const float* __restrict__ Wih,
                                               const float* __restrict__ Whh,
                                               const float* __restrict__ bih,
                                               const float* __restrict__ bhh,
                                               float* __restrict__ gates, int t) {
  const int lane = threadIdx.x;
  const int lmod = lane & 15, lhalf = lane >> 4;
  const int j0 = (blockIdx.x * 4 + threadIdx.y) * 16;   // output-unit tile base
  const int mBase = blockIdx.y * 16;                    // batch tile base
  const int aRow = mBase + lmod;
  const int wRow = j0 + lmod;

  v8f acc[4] = {};
  const float* xt = x_seq + (size_t)t * BATCH * IN_DIM;
  gemm_seg(xt, IN_DIM, Wih, CTRL_IN, 0,       IN_DIM, aRow, wRow, lhalf, acc); // x part
  gemm_seg(rv, RW_DIM, Wih, CTRL_IN, IN_DIM,  RW_DIM, aRow, wRow, lhalf, acc); // read_vec part
  gemm_seg(h,  H_DIM,  Whh, H_DIM,   0,       H_DIM,  aRow, wRow, lhalf, acc); // h part

  v8f tot = acc[0] + acc[1] + acc[2] + acc[3];
  const int j = j0 + lmod;
  const float bb = bih[j] + bhh[j];
#pragma unroll
  for (int r = 0; r < 8; ++r) {
    const int m = mBase + r + 8 * lhalf;                // C/D layout (ISA 7.12.2)
    gates[(size_t)m * G4H + j] = tot[r] + bb;
  }
}

// ---------------------------------------------------------------------------
// Kernel 2: LSTM elementwise update (i,f,g,o split along axis 1)
// ---------------------------------------------------------------------------
__global__ void k_lstm(const float* __restrict__ gates,
                       float* __restrict__ c, float* __restrict__ h) {
  const int idx = blockIdx.x * blockDim.x + threadIdx.x;  // 0 .. B*H-1
  const int m = idx >> 9, j = idx & (H_DIM - 1);
  const float* g = gates + (size_t)m * G4H;
  const float gi = g[j], gf = g[H_DIM + j], gg = g[2 * H_DIM + j], go = g[3 * H_DIM + j];
  const float cn = sigmoidf(gf) * c[idx] + sigmoidf(gi) * tanhf(gg);
  c[idx] = cn;
  h[idx] = sigmoidf(go) * tanhf(cn);
}

// ---------------------------------------------------------------------------
// Kernel 3: out = h@Wout^T + bout  (-> d_out[t])  and keys = h@Wkey^T + bkey
// 28 column tiles (8 Wout + 20 Wkey). block=(32,4), grid=(7,4).
// ---------------------------------------------------------------------------
__global__ __launch_bounds__(128) void k_outkeys(const float* __restrict__ h,
                                                 const float* __restrict__ Wout,
                                                 const float* __restrict__ bout,
                                                 const float* __restrict__ Wkey,
                                                 const float* __restrict__ bkey,
                                                 float* __restrict__ out,
                                                 float* __restrict__ keys, int t) {
  const int lane = threadIdx.x;
  const int lmod = lane & 15, lhalf = lane >> 4;
  const int jj0 = (blockIdx.x * 4 + threadIdx.y) * 16;  // 0..447
  const int mBase = blockIdx.y * 16;
  const bool isOut = (jj0 < OUT_DIM);                   // uniform per wave
  const int jl = isOut ? jj0 : (jj0 - OUT_DIM);
  const float* W = isOut ? Wout : Wkey;
  const float* bias = isOut ? bout : bkey;

  v8f acc[4] = {};
  gemm_seg(h, H_DIM, W, H_DIM, 0, H_DIM, mBase + lmod, jl + lmod, lhalf, acc);
  v8f tot = acc[0] + acc[1] + acc[2] + acc[3];
  const float bb = bias[jl + lmod];
#pragma unroll
  for (int r = 0; r < 8; ++r) {
    const int m = mBase + r + 8 * lhalf;
    const float v = tot[r] + bb;
    if (isOut) out[((size_t)t * BATCH + m) * OUT_DIM + jj0 + lmod] = v;
    else       keys[(size_t)m * KEY_DIM + jl + lmod] = v;
  }
}

// ---------------------------------------------------------------------------
// Kernel 4: fused memory subsystem — one workgroup (256 thr / 8 waves) per b.
// argmin(usage) -> cosine sim (sim kept in 32KB LDS) -> softmax ->
// read_vec accumulation + least-used write + usage/readsum update.
// All reductions fixed-order (deterministic).
// ---------------------------------------------------------------------------
__global__ __launch_bounds__(256) void k_memory(float* __restrict__ M,
                                                float* __restrict__ usage,
                                                float* __restrict__ rwsum,
                                                float* __restrict__ read_vec,
                                                const float* __restrict__ keys,
                                                const float* __restrict__ alpha_p,
                                                const float* __restrict__ gamma_p) {
  __shared__ float s_sim[R_HEADS][N_SLOTS];   // sim, then exp(sim - max)
  __shared__ float s_key[R_HEADS + 1][W_DIM];
  __shared__ float s_invk[R_HEADS];
  __shared__ float s_red[256];
  __shared__ int   s_redi[256];
  __shared__ float s_wred[R_HEADS][8];        // per-wave partials (max / sums)
  __shared__ float s_rmax[R_HEADS];
  __shared__ float s_den[R_HEADS];
  __shared__ float s_rvp[8][R_HEADS][W_DIM];  // per-wave read-vector partials
  __shared__ int   s_lu;

  const int b = blockIdx.x;
  const int tid = threadIdx.x;
  const int lane = tid & 31;
  const int wave = tid >> 5;

  float* Mb = M + (size_t)b * N_SLOTS * W_DIM;
  float* ub = usage + (size_t)b * N_SLOTS;
  float* rb = rwsum + (size_t)b * N_SLOTS;

  // stage keys; invk = 1/(||key_r|| + eps)
  for (int i = tid; i < KEY_DIM; i += 256)
    s_key[i / W_DIM][i % W_DIM] = keys[(size_t)b * KEY_DIM + i];
  __syncthreads();
  if (tid < R_HEADS) {
    float s = 0.f;
    for (int w = 0; w < W_DIM; ++w) { const float k = s_key[tid][w]; s += k * k; }
    s_invk[tid] = 1.f / (sqrtf(s) + EPSF);
  }

  // argmin(usage) (first-min semantics, matching jnp.argmin)
  float mv = 3.4e38f; int mi = N_SLOTS;
  for (int n = tid; n < N_SLOTS; n += 256) {
    const float u = ub[n];
    if (u < mv) { mv = u; mi = n; }
  }
  s_red[tid] = mv; s_redi[tid] = mi;
  __syncthreads();
  for (int off = 128; off > 0; off >>= 1) {
    if (tid < off) {
      const float v2 = s_red[tid + off]; const int i2 = s_redi[tid + off];
      if (v2 < s_red[tid] || (v2 == s_red[tid] && i2 < s_redi[tid])) {
        s_red[tid] = v2; s_redi[tid] = i2;
      }
    }
    __syncthreads();
  }
  if (tid == 0) s_lu = s_redi[0];
  __syncthreads();

  // Phase B: sim[r][n] = <M[n],key_r> / ((||M[n]||+eps)(||key_r||+eps))
  // one wave per contiguous 256-row chunk; 256B coalesced row loads.
  const int n0 = wave * (N_SLOTS / 8), n1 = n0 + (N_SLOTS / 8);
  float lm0 = -3.4e38f, lm1 = -3.4e38f, lm2 = -3.4e38f, lm3 = -3.4e38f;
  for (int n = n0; n < n1; ++n) {
    v2f m = *(const v2f*)(Mb + (size_t)n * W_DIM + 2 * lane);
    float nr = m.x * m.x + m.y * m.y;
    float d0 = m.x * s_key[0][2 * lane] + m.y * s_key[0][2 * lane + 1];
    float d1 = m.x * s_key[1][2 * lane] + m.y * s_key[1][2 * lane + 1];
    float d2 = m.x * s_key[2][2 * lane] + m.y * s_key[2][2 * lane + 1];
    float d3 = m.x * s_key[3][2 * lane] + m.y * s_key[3][2 * lane + 1];
#pragma unroll
    for (int off = 16; off > 0; off >>= 1) {
      nr += __shfl_xor(nr, off);
      d0 += __shfl_xor(d0, off); d1 += __shfl_xor(d1, off);
      d2 += __shfl_xor(d2, off); d3 += __shfl_xor(d3, off);
    }
    const float inv = 1.f / (sqrtf(nr) + EPSF);
    const float v0 = d0 * inv * s_invk[0], v1 = d1 * inv * s_invk[1];
    const float v2 = d2 * inv * s_invk[2], v3 = d3 * inv * s_invk[3];
    if (lane == 0) { s_sim[0][n] = v0; s_sim[1][n] = v1; s_sim[2][n] = v2; s_sim[3][n] = v3; }
    lm0 = fmaxf(lm0, v0); lm1 = fmaxf(lm1, v1); lm2 = fmaxf(lm2, v2); lm3 = fmaxf(lm3, v3);
  }
  if (lane == 0) { s_wred[0][wave] = lm0; s_wred[1][wave] = lm1; s_wred[2][wave] = lm2; s_wred[3][wave] = lm3; }
  __syncthreads();
  if (tid < R_HEADS) {
    float mx = -3.4e38f;
    for (int wv = 0; wv < 8; ++wv) mx = fmaxf(mx, s_wred[tid][wv]);
    s_rmax[tid] = mx;
  }
  __syncthreads();

  // softmax: exp pass + deterministic denominator (shfl tree + fixed-order sum)
  float ds0 = 0.f, ds1 = 0.f, ds2 = 0.f, ds3 = 0.f;
  for (int n = tid; n < N_SLOTS; n += 256) {
    float e;
    e = expf(s_sim[0][n] - s_rmax[0]); s_sim[0][n] = e; ds0 += e;
    e = expf(s_sim[1][n] - s_rmax[1]); s_sim[1][n] = e; ds1 += e;
    e = expf(s_sim[2][n] - s_rmax[2]); s_sim[2][n] = e; ds2 += e;
    e = expf(s_sim[3][n] - s_rmax[3]); s_sim[3][n] = e; ds3 += e;
  }
#pragma unroll
  for (int off = 16; off > 0; off >>= 1) {
    ds0 += __shfl_xor(ds0, off); ds1 += __shfl_xor(ds1, off);
    ds2 += __shfl_xor(ds2, off); ds3 += __shfl_xor(ds3, off);
  }
  __syncthreads();  // s_wred reuse
  if (lane == 0) { s_wred[0][wave] = ds0; s_wred[1][wave] = ds1; s_wred[2][wave] = ds2; s_wred[3][wave] = ds3; }
  // zero read-vector partials
  {
    float* z = &s_rvp[0][0][0];
    for (int i = tid; i < 8 * R_HEADS * W_DIM; i += 256) z[i] = 0.f;
  }
  __syncthreads();
  if (tid < R_HEADS) {
    float s = 0.f;
    for (int wv = 0; wv < 8; ++wv) s += s_wred[tid][wv];
    s_den[tid] = s;
  }
  __syncthreads();

  // Phase C: read_vec accumulation + least-used write + usage/readsum update
  const float sa = sigmoidf(alpha_p[0]);
  const float gm = gamma_p[0];
  const int lu = s_lu;
  const float id0 = 1.f / s_den[0], id1 = 1.f / s_den[1];
  const float id2 = 1.f / s_den[2], id3 = 1.f / s_den[3];
  v2f wk; wk.x = s_key[R_HEADS][2 * lane]; wk.y = s_key[R_HEADS][2 * lane + 1];
  v2f rv0 = {0.f, 0.f}, rv1 = {0.f, 0.f}, rv2 = {0.f, 0.f}, rv3 = {0.f, 0.f};
  for (int n = n0; n < n1; ++n) {
    const size_t mo = (size_t)n * W_DIM + 2 * lane;
    v2f m = *(const v2f*)(Mb + mo);                 // pre-write row
    const float p0 = s_sim[0][n] * id0, p1 = s_sim[1][n] * id1;
    const float p2 = s_sim[2][n] * id2, p3 = s_sim[3][n] * id3;
    const float psum = p0 + p1 + p2 + p3;
    const float isLu = (n == lu) ? 1.f : 0.f;
    const float ww = sa * rb[n] + (1.f - sa) * isLu;  // uses OLD read_w sum
    const float keep = 1.f - isLu;
    v2f nm; nm.x = m.x * keep + ww * wk.x; nm.y = m.y * keep + ww * wk.y;
    *(v2f*)(Mb + mo) = nm;                          // M_new
    rv0.x += p0 * m.x; rv0.y += p0 * m.y;
    rv1.x += p1 * m.x; rv1.y += p1 * m.y;
    rv2.x += p2 * m.x; rv2.y += p2 * m.y;
    rv3.x += p3 * m.x; rv3.y += p3 * m.y;
    if (lane == 0) {
      ub[n] = gm * ub[n] + psum + ww;               // usage_new
      rb[n] = psum;                                 // carry read_w_new.sum(r)
    }
  }
  s_rvp[wave][0][2 * lane] = rv0.x; s_rvp[wave][0][2 * lane + 1] = rv0.y;
  s_rvp[wave][1][2 * lane] = rv1.x; s_rvp[wave][1][2 * lane + 1] = rv1.y;
  s_rvp[wave][2][2 * lane] = rv2.x; s_rvp[wave][2][2 * lane + 1] = rv2.y;
  s_rvp[wave][3][2 * lane] = rv3.x; s_rvp[wave][3][2 * lane + 1] = rv3.y;
  __syncthreads();
  {
    const int r = tid >> 6, w = tid & 63;           // tid == r*64+w
    float s = 0.f;
    for (int wv = 0; wv < 8; ++wv) s += s_rvp[wv][r][w];   // fixed order
    read_vec[(size_t)b * RW_DIM + tid] = s;
  }
}

// ---------------------------------------------------------------------------
// state init: zero h,c,read_vec,usage,rwsum ; M = 1e-6
// ---------------------------------------------------------------------------
__global__ void k_init(float* __restrict__ ws) {
  const size_t i0 = (size_t)blockIdx.x * blockDim.x + threadIdx.x;
  const size_t stride = (size_t)gridDim.x * blockDim.x;
  for (size_t x = i0; x < (size_t)OFF_GATES; x += stride) ws[x] = 0.f;
  for (size_t x = i0; x < (size_t)M_ELEMS; x += stride) ws[OFF_M + x] = 1e-6f;
}

extern "C" void kernel_launch(void* const* d_in, const int* in_sizes, int n_in,
                              void* d_out, int out_size, void* d_ws, size_t ws_size,
                              hipStream_t stream) {
  (void)in_sizes; (void)n_in; (void)out_size; (void)ws_size;  // WS_TOTAL*4 ≈ 35.5 MB needed
  const float* x_seq = (const float*)d_in[0];
  const float* Wih   = (const float*)d_in[1];
  const float* Whh   = (const float*)d_in[2];
  const float* bih   = (const float*)d_in[3];
  const float* bhh   = (const float*)d_in[4];
  const float* Wout  = (const float*)d_in[5];
  const float* bout  = (const float*)d_in[6];
  const float* Wkey  = (const float*)d_in[7];
  const float* bkey  = (const float*)d_in[8];
  const float* alpha = (const float*)d_in[9];
  const float* gamma = (const float*)d_in[10];
  float* out = (float*)d_out;
  float* ws  = (float*)d_ws;

  float* h     = ws + OFF_H;
  float* c     = ws + OFF_C;
  float* rv    = ws + OFF_RV;
  float* usage = ws + OFF_USAGE;
  float* rwsum = ws + OFF_RWSUM;
  float* gates = ws + OFF_GATES;
  float* keys  = ws + OFF_KEYS;
  float* M     = ws + OFF_M;

  k_init<<<2048, 256, 0, stream>>>(ws);

  for (int t = 0; t < T_STEPS; ++t) {
    k_gates<<<dim3(32, 4), dim3(32, 4), 0, stream>>>(x_seq, rv, h, Wih, Whh, bih, bhh, gates, t);
    k_lstm<<<(BATCH * H_DIM) / 256, 256, 0, stream>>>(gates, c, h);
    k_outkeys<<<dim3(7, 4), dim3(32, 4), 0, stream>>>(h, Wout, bout, Wkey, bkey, out, keys, t);
    k_memory<<<BATCH, 256, 0, stream>>>(M, usage, rwsum, rv, keys, alpha, gamma);
  }
}